// LightGCN_51067161150287
// MI455X (gfx1250) — compile-verified
//
#include <hip/hip_runtime.h>
#include <hip/hip_bf16.h>

#define NUM_USERS 100000
#define NUM_ITEMS 50000
#define N_NODES   (NUM_USERS + NUM_ITEMS)
#define N_EDGES   3200000
#define EMB_DIM   128
#define N_LAYERS  3
#define N_QUERY   2048

typedef float v2f __attribute__((ext_vector_type(2)));
typedef float v8f __attribute__((ext_vector_type(8)));

// Native CDNA5 no-return f32 atomic add (L2-side, STOREcnt-tracked).
// s_endpgm performs an implicit wait-idle, and kernel boundaries provide
// device-scope visibility for the next kernel's reads.
__device__ __forceinline__ void atom_add_f32(float* p, float v) {
    asm volatile("global_atomic_add_f32 %0, %1, off"
                 :
                 : "v"(p), "v"(v)
                 : "memory");
}

// ---------------------------------------------------------------------------
// emb[0:NUM_USERS] = user_emb ; emb[NUM_USERS:] = item_emb  (float4 copies)
// ---------------------------------------------------------------------------
__global__ void k_concat(const float* __restrict__ ue,
                         const float* __restrict__ ie,
                         float* __restrict__ emb) {
    const long long i = (long long)blockIdx.x * blockDim.x + threadIdx.x; // float4 index
    const long long nu4  = (long long)NUM_USERS * (EMB_DIM / 4);
    const long long tot4 = (long long)N_NODES   * (EMB_DIM / 4);
    if (i >= tot4) return;
    float4 v = (i < nu4) ? ((const float4*)ue)[i] : ((const float4*)ie)[i - nu4];
    ((float4*)emb)[i] = v;
}

__global__ void k_zero(float4* __restrict__ p, long long n4) {
    const long long i = (long long)blockIdx.x * blockDim.x + threadIdx.x;
    if (i < n4) p[i] = make_float4(0.f, 0.f, 0.f, 0.f);
}

// ---------------------------------------------------------------------------
// SpMM scatter: one wave per edge (grid-stride over waves).
// Edge index forced into an SGPR via readfirstlane so the row/col/val loads
// become scalar (SMEM, KMcnt) instead of per-lane VMEM. 32 lanes * float4 =
// 512B coalesced gather of the source row, then 4 no-return f32 atomics per
// lane into the destination row (both tables L2-resident: 153.6MB < 192MB L2).
// ---------------------------------------------------------------------------
__global__ void k_scatter(const int*   __restrict__ erow,
                          const int*   __restrict__ ecol,
                          const float* __restrict__ eval,
                          const float* __restrict__ src,
                          float*       __restrict__ dst) {
    const int lane = threadIdx.x & 31;
    const int wave0  = (int)(((long long)blockIdx.x * blockDim.x + threadIdx.x) >> 5);
    const int nwaves = (int)(((long long)gridDim.x * blockDim.x) >> 5);
    for (int e = wave0; e < N_EDGES; e += nwaves) {
        const int eu = __builtin_amdgcn_readfirstlane(e); // wave-uniform -> SGPR
        const int   r = erow[eu];
        const int   c = ecol[eu];
        const float w = eval[eu];
        const float4 s = *(const float4*)(src + (long long)c * EMB_DIM + lane * 4);
        float* d = dst + (long long)r * EMB_DIM + lane * 4;
        atom_add_f32(d + 0, w * s.x);
        atom_add_f32(d + 1, w * s.y);
        atom_add_f32(d + 2, w * s.z);
        atom_add_f32(d + 3, w * s.w);
    }
}

// ---------------------------------------------------------------------------
// Gather the 2048 selected user rows + 2048 selected item rows out of the
// current layer embedding and accumulate into small (2 MB) buffers.
// init==1 : store (first contribution, also initializes workspace)
// init==0 : add
// ---------------------------------------------------------------------------
__global__ void k_gather_acc(const float* __restrict__ emb,
                             const int*   __restrict__ users,
                             const int*   __restrict__ items,
                             float* __restrict__ uacc,
                             float* __restrict__ iacc,
                             int init) {
    const int lane = threadIdx.x & 31;
    const int wave = (int)(((long long)blockIdx.x * blockDim.x + threadIdx.x) >> 5);
    if (wave >= 2 * N_QUERY) return;
    long long srow;
    float* d;
    if (wave < N_QUERY) {
        srow = users[wave];
        d = uacc + (long long)wave * EMB_DIM;
    } else {
        srow = (long long)NUM_USERS + items[wave - N_QUERY];
        d = iacc + (long long)(wave - N_QUERY) * EMB_DIM;
    }
    const float4 s = *(const float4*)(emb + srow * EMB_DIM + lane * 4);
    float4* dp = (float4*)(d + lane * 4);
    if (init) {
        *dp = s;
    } else {
        float4 o = *dp;
        o.x += s.x; o.y += s.y; o.z += s.z; o.w += s.w;
        *dp = o;
    }
}

// ---------------------------------------------------------------------------
// Final stage: gamma[i] = dot(uacc[i], iacc[i]) / 16 via V_WMMA_F32_16X16X4_F32.
// One wave handles 16 (user,item) pairs: A = 16 user rows (16xK), B = 16 item
// columns (Kx16); diagonal of D = the 16 dot products.
// A/B f32 layout (16x4): lanes 0-15 hold K={0,1}, lanes 16-31 hold K={2,3},
// so lane l loads row (base + (l&15)) at K-offset ((l>>4)*2).
// D layout: vgpr r / lanes 0-15 -> M=r, N=lane ; lanes 16-31 -> M=8+r, N=lane-16.
// Diagonal: (lane l<8, d[l]) and (lane 24+r, d[r]).
// ---------------------------------------------------------------------------
__global__ void k_dot_wmma(const float* __restrict__ uacc,
                           const float* __restrict__ iacc,
                           float* __restrict__ out) {
    const int lane = threadIdx.x & 31;
    const int wave = (int)(((long long)blockIdx.x * blockDim.x + threadIdx.x) >> 5);
    if (wave >= N_QUERY / 16) return;
    const int base  = wave * 16;
    const int row   = base + (lane & 15);
    const int khalf = (lane >> 4) * 2; // 0 or 2

    const float* __restrict__ up = uacc + (long long)row * EMB_DIM + khalf;
    const float* __restrict__ ip = iacc + (long long)row * EMB_DIM + khalf;

    v8f c = {0.f, 0.f, 0.f, 0.f, 0.f, 0.f, 0.f, 0.f};
#pragma unroll
    for (int k0 = 0; k0 < EMB_DIM; k0 += 4) {
        v2f a = {up[k0], up[k0 + 1]};
        v2f b = {ip[k0], ip[k0 + 1]};
        c = __builtin_amdgcn_wmma_f32_16x16x4_f32(false, a, false, b,
                                                  (short)0, c, false, false);
    }

    float g = 0.0f;
#pragma unroll
    for (int r = 0; r < 8; ++r) {
        if (lane == r)      g = c[r];
        if (lane == r + 24) g = c[r];
    }
    const float scale = 1.0f / 16.0f; // (acc/4) . (acc/4)
    if (lane < 8)        out[base + lane]      = g * scale;
    else if (lane >= 24) out[base + lane - 16] = g * scale;
}

// ---------------------------------------------------------------------------
// Host-side orchestration.
// Workspace layout (floats):
//   embA : N_NODES*128   (76.8 MB)
//   embB : N_NODES*128   (76.8 MB)
//   uacc : 2048*128      (1 MB)
//   iacc : 2048*128      (1 MB)
// ---------------------------------------------------------------------------
extern "C" void kernel_launch(void* const* d_in, const int* in_sizes, int n_in,
                              void* d_out, int out_size, void* d_ws, size_t ws_size,
                              hipStream_t stream) {
    const float* user_emb = (const float*)d_in[0];
    const float* item_emb = (const float*)d_in[1];
    const float* edge_val = (const float*)d_in[2];
    const int*   edge_row = (const int*)d_in[3];
    const int*   edge_col = (const int*)d_in[4];
    const int*   users    = (const int*)d_in[5];
    const int*   items    = (const int*)d_in[6];
    float*       out      = (float*)d_out;

    float* ws   = (float*)d_ws;
    float* embA = ws;
    float* embB = embA + (size_t)N_NODES * EMB_DIM;
    float* uacc = embB + (size_t)N_NODES * EMB_DIM;
    float* iacc = uacc + (size_t)N_QUERY * EMB_DIM;

    const long long tot4 = (long long)N_NODES * (EMB_DIM / 4);
    const int gcat = (int)((tot4 + 255) / 256);
    const int ggat = (2 * N_QUERY * 32 + 255) / 256;   // 4096 waves
    const int gdot = (N_QUERY / 16 * 32) / 256;        // 128 waves, exact

    k_concat<<<gcat, 256, 0, stream>>>(user_emb, item_emb, embA);
    k_gather_acc<<<ggat, 256, 0, stream>>>(embA, users, items, uacc, iacc, 1);

    float* cur = embA;
    float* nxt = embB;
    for (int l = 0; l < N_LAYERS; ++l) {
        k_zero<<<gcat, 256, 0, stream>>>((float4*)nxt, tot4);
        k_scatter<<<8192, 256, 0, stream>>>(edge_row, edge_col, edge_val, cur, nxt);
        k_gather_acc<<<ggat, 256, 0, stream>>>(nxt, users, items, uacc, iacc, 0);
        float* t = cur; cur = nxt; nxt = t;
    }

    k_dot_wmma<<<gdot, 256, 0, stream>>>(uacc, iacc, out);
}